// GATNet_76227079569578
// MI455X (gfx1250) — compile-verified
//
#include <hip/hip_runtime.h>

// ---------------- problem constants (match reference) ----------------
#define F_IN   256
#define HEADS  6
#define CDIM   128           // C1 == OUT_C == 128
#define FDIM   768           // HEADS * CDIM
#define OUT_C  128
#define NEG_SLOPE 0.2f
#define BN_EPS    1e-5f

typedef __attribute__((ext_vector_type(2))) float v2f;
typedef __attribute__((ext_vector_type(8))) float v8f;

// Monotone bijection float -> uint so unsigned atomicMax == float max.
__device__ __forceinline__ unsigned fmap(float f) {
    unsigned u = __float_as_uint(f);
    return (u & 0x80000000u) ? ~u : (u | 0x80000000u);
}
__device__ __forceinline__ float funmap(unsigned u) {
    u = (u & 0x80000000u) ? (u & 0x7FFFFFFFu) : ~u;
    return __uint_as_float(u);
}
#define FMAP_NEG_INF 0x007FFFFFu   // fmap(-inf)

#define WMMA_F32(a, b, c) \
    __builtin_amdgcn_wmma_f32_16x16x4_f32(false, (a), false, (b), (short)0, (c), false, false)

// ---------------------------------------------------------------------
// fp32 GEMM on the WMMA path: C[M,Nc] = A[M,K] @ B[K,Nc] (+bias)
// One wave computes a 32(M) x 64(N) strip (2 M-subtiles x 4 N-tiles =
// 8 accumulators); every B fetch is reused by both M-subtiles, so the
// steady-state inner loop is 10 VMEM loads : 8 v_wmma_f32_16x16x4_f32.
// Requires Nc%64==0, K%4==0.  M handled to 16-row granularity: the
// second M-subtile of the last tile is uniformly clamped/skipped
// (WMMA needs EXEC all-ones, so we clamp addresses, never lanes).
//
// fp32 WMMA lane layout (ISA 7.12.2):
//   A 16x4 : lanes 0-15 -> M=lane, K={0,1}; lanes 16-31 -> M=lane-16, K={2,3}
//   B 4x16 : lanes 0-15 -> N=lane, K={0,1}; lanes 16-31 -> N=lane-16, K={2,3}
//   C 16x16: vgpr r, lanes 0-15 -> M=r; lanes 16-31 -> M=8+r; N=lane&15
// ---------------------------------------------------------------------
__global__ __launch_bounds__(32)
void wmma_gemm_f32(const float* __restrict__ A, const float* __restrict__ B,
                   float* __restrict__ C, const float* __restrict__ bias,
                   int M, int K, int Nc)
{
    const int lane  = threadIdx.x & 31;
    const int nN64  = Nc >> 6;
    const int m0    = (blockIdx.x / nN64) * 32;
    const int n0    = (blockIdx.x % nN64) * 64;
    const int lrow  = lane & 15;          // M (for A) / N (for B,C) within tile
    const int khalf = (lane >> 4) << 1;   // 0 for lanes 0-15, 2 for 16-31
    const bool hasM1 = (m0 + 16) < M;     // wave-uniform

    v8f acc00 = {}, acc01 = {}, acc02 = {}, acc03 = {};
    v8f acc10 = {}, acc11 = {}, acc12 = {}, acc13 = {};

    const float* arow0 = A + (size_t)(m0 + lrow) * K;
    const float* arow1 = A + (size_t)(hasM1 ? (m0 + 16 + lrow) : (m0 + lrow)) * K;

    for (int k = 0; k < K; k += 4) {
        v2f a0, a1;
        a0.x = arow0[k + khalf];
        a0.y = arow0[k + khalf + 1];
        a1.x = arow1[k + khalf];
        a1.y = arow1[k + khalf + 1];

        const float* brow0 = B + (size_t)(k + khalf) * Nc + n0 + lrow;
        const float* brow1 = brow0 + Nc;
        v2f bv0, bv1, bv2, bv3;
        bv0.x = brow0[0];   bv0.y = brow1[0];
        bv1.x = brow0[16];  bv1.y = brow1[16];
        bv2.x = brow0[32];  bv2.y = brow1[32];
        bv3.x = brow0[48];  bv3.y = brow1[48];

        acc00 = WMMA_F32(a0, bv0, acc00);
        acc01 = WMMA_F32(a0, bv1, acc01);
        acc02 = WMMA_F32(a0, bv2, acc02);
        acc03 = WMMA_F32(a0, bv3, acc03);
        acc10 = WMMA_F32(a1, bv0, acc10);
        acc11 = WMMA_F32(a1, bv1, acc11);
        acc12 = WMMA_F32(a1, bv2, acc12);
        acc13 = WMMA_F32(a1, bv3, acc13);
    }

    float bb0 = 0.f, bb1 = 0.f, bb2 = 0.f, bb3 = 0.f;
    if (bias) {
        bb0 = bias[n0 + lrow];
        bb1 = bias[n0 + 16 + lrow];
        bb2 = bias[n0 + 32 + lrow];
        bb3 = bias[n0 + 48 + lrow];
    }
    const int rbase = (lane >> 4) * 8;
    #pragma unroll
    for (int r = 0; r < 8; ++r) {
        float* crow = C + (size_t)(m0 + rbase + r) * Nc + n0 + lrow;
        crow[0]  = acc00[r] + bb0;
        crow[16] = acc01[r] + bb1;
        crow[32] = acc02[r] + bb2;
        crow[48] = acc03[r] + bb3;
    }
    if (hasM1) {
        #pragma unroll
        for (int r = 0; r < 8; ++r) {
            float* crow = C + (size_t)(m0 + 16 + rbase + r) * Nc + n0 + lrow;
            crow[0]  = acc10[r] + bb0;
            crow[16] = acc11[r] + bb1;
            crow[32] = acc12[r] + bb2;
            crow[48] = acc13[r] + bb3;
        }
    }
}

// ---------------------------------------------------------------------
// a_s[n,h] = dot(hx[n,h,:], att_s[h,:]) ; same for a_d.  One wave per
// (node, head): 32 lanes x float4 covers CDIM=128.
// ---------------------------------------------------------------------
__global__ __launch_bounds__(256)
void attn_scores(const float* __restrict__ hx,
                 const float* __restrict__ att_s, const float* __restrict__ att_d,
                 float* __restrict__ a_s, float* __restrict__ a_d, int nNodes)
{
    int gid  = blockIdx.x * blockDim.x + threadIdx.x;
    int wave = gid >> 5;
    int lane = gid & 31;
    if (wave >= nNodes * HEADS) return;
    int n = wave / HEADS, h = wave % HEADS;

    const float4* hv = (const float4*)(hx + (size_t)n * FDIM + h * CDIM);
    const float4* sv = (const float4*)(att_s + h * CDIM);
    const float4* dv = (const float4*)(att_d + h * CDIM);
    float4 x = hv[lane], s = sv[lane], d = dv[lane];
    float accs = x.x * s.x + x.y * s.y + x.z * s.z + x.w * s.w;
    float accd = x.x * d.x + x.y * d.y + x.z * d.z + x.w * d.w;
    #pragma unroll
    for (int off = 16; off > 0; off >>= 1) {
        accs += __shfl_down(accs, off, 32);
        accd += __shfl_down(accd, off, 32);
    }
    if (lane == 0) { a_s[wave] = accs; a_d[wave] = accd; }
}

__global__ void fill_u32(unsigned* __restrict__ p, unsigned v, size_t n)
{
    size_t i = (size_t)blockIdx.x * blockDim.x + threadIdx.x;
    if (i < n) p[i] = v;
}

// e = leaky_relu(a_s[src]+a_d[dst]); store e; segment-max into mbuf[dst].
__global__ __launch_bounds__(256)
void edge_scores_max(const int* __restrict__ adj,
                     const float* __restrict__ a_s, const float* __restrict__ a_d,
                     float* __restrict__ ebuf, unsigned* __restrict__ mbuf,
                     int E, int nTot)
{
    int e = blockIdx.x * blockDim.x + threadIdx.x;
    if (e >= nTot) return;
    int src, dst;
    if (e < E) { src = adj[e]; dst = adj[E + e]; } else { src = dst = e - E; }
    #pragma unroll
    for (int h = 0; h < HEADS; ++h) {
        float v = a_s[src * HEADS + h] + a_d[dst * HEADS + h];
        v = (v > 0.f) ? v : v * NEG_SLOPE;
        ebuf[(size_t)e * HEADS + h] = v;
        atomicMax(&mbuf[dst * HEADS + h], fmap(v));
    }
}

// e <- exp(e - m[dst]); segment-sum into sbuf[dst].
__global__ __launch_bounds__(256)
void edge_exp_sum(const int* __restrict__ adj, float* __restrict__ ebuf,
                  const unsigned* __restrict__ mbuf, float* __restrict__ sbuf,
                  int E, int nTot)
{
    int e = blockIdx.x * blockDim.x + threadIdx.x;
    if (e >= nTot) return;
    int dst = (e < E) ? adj[E + e] : e - E;
    #pragma unroll
    for (int h = 0; h < HEADS; ++h) {
        float m = funmap(mbuf[dst * HEADS + h]);
        float p = __expf(ebuf[(size_t)e * HEADS + h] - m);
        ebuf[(size_t)e * HEADS + h] = p;
        atomicAdd(&sbuf[dst * HEADS + h], p);
    }
}

// out[dst,h,:] += hx[src,h,:] * (e / s[dst,h]).  One wave per (edge, head).
__global__ __launch_bounds__(256)
void edge_aggregate(const int* __restrict__ adj, const float* __restrict__ hx,
                    const float* __restrict__ ebuf, const float* __restrict__ sbuf,
                    float* __restrict__ out, int E, int nTot)
{
    long gid  = (long)blockIdx.x * blockDim.x + threadIdx.x;
    long wave = gid >> 5;
    int  lane = (int)(gid & 31);
    if (wave >= (long)nTot * HEADS) return;
    int e = (int)(wave / HEADS), h = (int)(wave % HEADS);
    int src, dst;
    if (e < E) { src = adj[e]; dst = adj[E + e]; } else { src = dst = e - E; }

    float alpha = ebuf[(size_t)e * HEADS + h] / sbuf[dst * HEADS + h];
    const float4* hv = (const float4*)(hx + (size_t)src * FDIM + h * CDIM);
    float4 v = hv[lane];
    float* o = out + (size_t)dst * FDIM + h * CDIM + lane * 4;
    atomicAdd(o + 0, v.x * alpha);
    atomicAdd(o + 1, v.y * alpha);
    atomicAdd(o + 2, v.z * alpha);
    atomicAdd(o + 3, v.w * alpha);
}

__global__ __launch_bounds__(256)
void bias_relu(float* __restrict__ x, const float* __restrict__ bias, size_t total)
{
    size_t i = (size_t)blockIdx.x * blockDim.x + threadIdx.x;
    if (i >= total) return;
    float v = x[i] + bias[i % FDIM];
    x[i] = v > 0.f ? v : 0.f;
}

// One block per column: biased mean/var over rows (matches jnp.var ddof=0).
__global__ __launch_bounds__(256)
void bn_stats(const float* __restrict__ x, float* __restrict__ mean,
              float* __restrict__ rstd, int rows)
{
    int col = blockIdx.x;
    float s = 0.f, ss = 0.f;
    for (int r = threadIdx.x; r < rows; r += 256) {
        float v = x[(size_t)r * FDIM + col];
        s += v; ss += v * v;
    }
    __shared__ float sh[256], sh2[256];
    sh[threadIdx.x] = s; sh2[threadIdx.x] = ss;
    __syncthreads();
    for (int o = 128; o > 0; o >>= 1) {
        if (threadIdx.x < o) {
            sh[threadIdx.x]  += sh[threadIdx.x + o];
            sh2[threadIdx.x] += sh2[threadIdx.x + o];
        }
        __syncthreads();
    }
    if (threadIdx.x == 0) {
        float mu  = sh[0] / rows;
        float var = sh2[0] / rows - mu * mu;
        mean[col] = mu;
        rstd[col] = rsqrtf(var + BN_EPS);
    }
}

__global__ __launch_bounds__(256)
void bn_apply(float* __restrict__ x, const float* __restrict__ mean,
              const float* __restrict__ rstd, const float* __restrict__ gamma,
              const float* __restrict__ beta, size_t total)
{
    size_t i = (size_t)blockIdx.x * blockDim.x + threadIdx.x;
    if (i >= total) return;
    int col = (int)(i % FDIM);
    x[i] = (x[i] - mean[col]) * rstd[col] * gamma[col] + beta[col];
}

__global__ __launch_bounds__(256)
void pool_max(const float* __restrict__ lin, const int* __restrict__ ibatch,
              unsigned* __restrict__ pooled, int rows)
{
    size_t i = (size_t)blockIdx.x * blockDim.x + threadIdx.x;
    if (i >= (size_t)rows * OUT_C) return;
    int n = (int)(i / OUT_C), c = (int)(i % OUT_C);
    atomicMax(&pooled[ibatch[n] * OUT_C + c], fmap(lin[i]));
}

__global__ __launch_bounds__(256)
void pool_final(const unsigned* __restrict__ pooled, float* __restrict__ out, int n)
{
    int i = blockIdx.x * blockDim.x + threadIdx.x;
    if (i < n) out[i] = funmap(pooled[i]);
}

// ---------------------------------------------------------------------
extern "C" void kernel_launch(void* const* d_in, const int* in_sizes, int n_in,
                              void* d_out, int out_size, void* d_ws, size_t ws_size,
                              hipStream_t stream)
{
    const float* x_in   = (const float*)d_in[0];
    const int*   adj    = (const int*)d_in[1];
    const int*   ibatch = (const int*)d_in[2];
    const float* W1     = (const float*)d_in[3];
    const float* att_s1 = (const float*)d_in[4];
    const float* att_d1 = (const float*)d_in[5];
    const float* b1     = (const float*)d_in[6];
    const float* g1     = (const float*)d_in[7];
    const float* be1    = (const float*)d_in[8];
    const float* W2     = (const float*)d_in[9];
    const float* att_s2 = (const float*)d_in[10];
    const float* att_d2 = (const float*)d_in[11];
    const float* b2     = (const float*)d_in[12];
    const float* g2     = (const float*)d_in[13];
    const float* be2    = (const float*)d_in[14];
    const float* linW   = (const float*)d_in[15];
    const float* linb   = (const float*)d_in[16];
    float* out = (float*)d_out;

    const int N    = in_sizes[0] / F_IN;   // 10000
    const int E    = in_sizes[1] / 2;      // 100000
    const int Etot = E + N;                // + self loops
    const int B    = out_size / OUT_C;     // 64
    const int mT   = (N + 31) / 32;        // 32-row GEMM tiles (last one half)

    // ---- workspace carve-up (fp32 units) ----
    float*    hx   = (float*)d_ws;                       // [N, FDIM]
    float*    agg  = hx  + (size_t)N * FDIM;             // [N, FDIM]
    float*    a_s  = agg + (size_t)N * FDIM;             // [N, HEADS]
    float*    a_d  = a_s + (size_t)N * HEADS;            // [N, HEADS]
    unsigned* mbuf = (unsigned*)(a_d + (size_t)N * HEADS); // [N, HEADS]
    float*    sbuf = (float*)(mbuf + (size_t)N * HEADS); // [N, HEADS]
    float*    ebuf = sbuf + (size_t)N * HEADS;           // [Etot, HEADS]
    float*    mean = ebuf + (size_t)Etot * HEADS;        // [FDIM]
    float*    rstd = mean + FDIM;                        // [FDIM]
    float*    lin  = rstd + FDIM;                        // [N, OUT_C]
    unsigned* pooled = (unsigned*)(lin + (size_t)N * OUT_C); // [B, OUT_C]

    const size_t nh  = (size_t)N * HEADS;
    const size_t nf  = (size_t)N * FDIM;
    const long   ew  = (long)Etot * HEADS;   // aggregate waves

    auto layer = [&](const float* xin, int K, const float* W,
                     const float* asw, const float* adw,
                     const float* bias, const float* gamma, const float* beta) {
        wmma_gemm_f32<<<mT * (FDIM / 64), 32, 0, stream>>>(xin, W, hx, nullptr, N, K, FDIM);
        attn_scores<<<(int)((nh * 32 + 255) / 256), 256, 0, stream>>>(hx, asw, adw, a_s, a_d, N);
        fill_u32<<<(int)((nh + 255) / 256), 256, 0, stream>>>(mbuf, FMAP_NEG_INF, nh);
        fill_u32<<<(int)((nh + 255) / 256), 256, 0, stream>>>((unsigned*)sbuf, 0u, nh);
        fill_u32<<<(int)((nf + 255) / 256), 256, 0, stream>>>((unsigned*)agg, 0u, nf);
        edge_scores_max<<<(Etot + 255) / 256, 256, 0, stream>>>(adj, a_s, a_d, ebuf, mbuf, E, Etot);
        edge_exp_sum<<<(Etot + 255) / 256, 256, 0, stream>>>(adj, ebuf, mbuf, sbuf, E, Etot);
        edge_aggregate<<<(int)((ew * 32 + 255) / 256), 256, 0, stream>>>(adj, hx, ebuf, sbuf, agg, E, Etot);
        bias_relu<<<(int)((nf + 255) / 256), 256, 0, stream>>>(agg, bias, nf);
        bn_stats<<<FDIM, 256, 0, stream>>>(agg, mean, rstd, N);
        bn_apply<<<(int)((nf + 255) / 256), 256, 0, stream>>>(agg, mean, rstd, gamma, beta, nf);
    };

    layer(x_in, F_IN, W1, att_s1, att_d1, b1, g1, be1);   // -> agg (= x1)
    layer(agg,  FDIM, W2, att_s2, att_d2, b2, g2, be2);   // -> agg (= x2)

    // final linear (+bias fused) and per-graph max pool
    wmma_gemm_f32<<<mT * (OUT_C / 64), 32, 0, stream>>>(agg, linW, lin, linb, N, FDIM, OUT_C);
    fill_u32<<<(B * OUT_C + 255) / 256, 256, 0, stream>>>(pooled, FMAP_NEG_INF, (size_t)B * OUT_C);
    pool_max<<<(int)(((size_t)N * OUT_C + 255) / 256), 256, 0, stream>>>(lin, ibatch, pooled, N);
    pool_final<<<(B * OUT_C + 255) / 256, 256, 0, stream>>>(pooled, out, B * OUT_C);
}